// highConvHeteroBlock_72095321030699
// MI455X (gfx1250) — compile-verified
//
#include <hip/hip_runtime.h>

#define NP 20000
#define RR 15
#define EE 30000
#define NH 60000   // 3*NP

typedef __attribute__((ext_vector_type(2))) float    v2f;
typedef __attribute__((ext_vector_type(8))) float    v8f;
typedef __attribute__((ext_vector_type(4))) unsigned v4u;
typedef __attribute__((ext_vector_type(4))) int      v4i;
typedef __attribute__((ext_vector_type(8))) int      v8i;

__constant__ int c_tsrc[RR] = {0,1,2,0,1,2,0,1,2,0,0,1,1,2,2};
__constant__ int c_tdst[RR] = {0,1,2,0,1,2,0,1,2,1,2,0,2,0,1};
__constant__ int c_off[3]   = {0, 2*NP, NP};

// ---- float <-> monotonic uint key (for segment-max via atomicMax) ----
__device__ inline unsigned fkey(float f) {
  unsigned b = __float_as_uint(f);
  return (b & 0x80000000u) ? ~b : (b | 0x80000000u);
}
__device__ inline float funkey(unsigned k) {
  unsigned b = (k & 0x80000000u) ? (k & 0x7FFFFFFFu) : ~k;
  return __uint_as_float(b);
}

// ============================================================
// TDM: DMA a rows x width f32 tile (row-major, contiguous) from
// global memory into LDS. Issued by one wave; caller barriers.
// D# layout per CDNA5 ISA ch.8 (group0 128b, group1 256b).
// ============================================================
#if __has_builtin(__builtin_amdgcn_tensor_load_to_lds)
#define HAVE_TDM 1
__device__ inline void tdm_load_tile(void* ldsPtr, const void* gptr,
                                     unsigned width, unsigned rows) {
  // flat LDS address = {aperture_hi32, lds_offset}; low 32 bits = byte offset
  unsigned ldsoff = (unsigned)(unsigned long long)ldsPtr;
  unsigned long long ga = (unsigned long long)gptr;
  v4u g0 = { 1u,                                   // count=1 (valid), no gather
             ldsoff,                               // lds_addr (bytes)
             (unsigned)ga,                         // global_addr[31:0]
             (unsigned)((ga >> 32) & 0x01FFFFFFull) | 0x80000000u }; // [56:32] | type=2
  v8i g1 = { (int)(2u << 16),                      // data_size=2 (4B), wg_mask=0
             (int)(width << 16),                   // tensor_dim0[15:0]
             (int)(rows  << 16),                   // tensor_dim0 hi=0 | tensor_dim1 lo
             (int)(width << 16),                   // tensor_dim1 hi=0 | tile_dim0
             (int)rows,                            // tile_dim1 | tile_dim2=0
             (int)width,                           // tensor_dim0_stride[31:0]
             0, 0 };                               // stride hi / dim1_stride = 0
  v4i g2 = {0,0,0,0};
  v4i g3 = {0,0,0,0};
#if __clang_major__ >= 23
  v8i g4 = {0,0,0,0,0,0,0,0};
  __builtin_amdgcn_tensor_load_to_lds(g0, g1, g2, g3, g4, 0);
#else
  __builtin_amdgcn_tensor_load_to_lds(g0, g1, g2, g3, 0);
#endif
  __builtin_amdgcn_s_wait_tensorcnt(0);
}
#else
#define HAVE_TDM 0
#endif

// ============================================================
// GEMM: hs[r] = x[tsrc[r]] @ Wlin[r] + b ; hd[r] likewise (tdst)
// One wave computes a 16x64 strip via V_WMMA_F32_16X16X4_F32.
// ============================================================
__global__ void __launch_bounds__(256) k_lin(
    const float* __restrict__ xin, const float* __restrict__ Wl,
    const float* __restrict__ bl, float* __restrict__ hs, float* __restrict__ hd)
{
  __shared__ float ldsW[64*64];
  int sel   = blockIdx.y;
  int which = (sel >= RR) ? 1 : 0;
  int r     = which ? sel - RR : sel;

  const float* W = Wl + (size_t)r * 64 * 64;
#if HAVE_TDM
  if ((threadIdx.x >> 5) == 0) tdm_load_tile(ldsW, W, 64u, 64u);
#else
  for (int i = threadIdx.x * 4; i < 64*64; i += 1024)
    *(float4*)(ldsW + i) = *(const float4*)(W + i);
#endif
  __syncthreads();

  int wave = threadIdx.x >> 5;
  int lane = threadIdx.x & 31;
  int m    = lane & 15;       // row within 16-row tile / col within 16-col tile
  int kh   = lane >> 4;       // K half-select per ISA A/B layout

  int tile = blockIdx.x * 8 + wave;
  if (tile > NP/16 - 1) tile = NP/16 - 1;   // clamp: duplicate last tile (identical writes)
  int rowbase = tile * 16;
  int type = which ? c_tdst[r] : c_tsrc[r];

  const float* arow = xin + ((size_t)type*NP + rowbase + m) * 64;
  __builtin_prefetch(arow + 64*16, 0, 1);   // global_prefetch_b8, 16 rows ahead

  // A fragments: 16x4 f32 tile -> v2f per lane; VGPR0 holds K=0/2, VGPR1 K=1/3
  v2f afrag[16];
  #pragma unroll
  for (int kk = 0; kk < 16; ++kk)
    afrag[kk] = *(const v2f*)(arow + kk*4 + kh*2);

  const float* bias = bl + r*64;
  float* outp = (which ? hd : hs) + ((size_t)r*NP + rowbase) * 64;

  #pragma unroll
  for (int t = 0; t < 4; ++t) {
    v8f acc = {};
    int col = t*16 + m;
    #pragma unroll
    for (int kk = 0; kk < 16; ++kk) {
      int kc = kk*4 + kh*2;
      v2f b;
      b.x = ldsW[kc*64 + col];
      b.y = ldsW[(kc+1)*64 + col];
      acc = __builtin_amdgcn_wmma_f32_16x16x4_f32(false, afrag[kk], false, b,
                                                  (short)0, acc, false, false);
    }
    float bv = bias[col];
    #pragma unroll
    for (int i = 0; i < 8; ++i)
      outp[(size_t)(i + kh*8)*64 + col] = acc[i] + bv;   // C layout: M = i + 8*laneHalf
  }
}

// ============================================================
// q/k/v/skip projections of hx (60000x64) via WMMA f32
// ============================================================
__global__ void __launch_bounds__(256) k_qkv(
    const float* __restrict__ hx,
    const float* __restrict__ Wq, const float* __restrict__ bq,
    const float* __restrict__ Wk, const float* __restrict__ bk,
    const float* __restrict__ Wv, const float* __restrict__ bv,
    const float* __restrict__ Ws, const float* __restrict__ bs,
    float* __restrict__ q, float* __restrict__ k,
    float* __restrict__ v, float* __restrict__ skip)
{
  __shared__ float ldsW[64*128];
  int which = blockIdx.y;
  const float* W; const float* bias; float* outp; int ldw;
  switch (which) {
    case 0:  W = Wq; bias = bq; outp = q;    ldw = 128; break;
    case 1:  W = Wk; bias = bk; outp = k;    ldw = 128; break;
    case 2:  W = Wv; bias = bv; outp = v;    ldw = 128; break;
    default: W = Ws; bias = bs; outp = skip; ldw = 64;  break;
  }
#if HAVE_TDM
  if ((threadIdx.x >> 5) == 0) tdm_load_tile(ldsW, W, (unsigned)ldw, 64u);
#else
  for (int i = threadIdx.x * 4; i < 64 * ldw; i += 1024)
    *(float4*)(ldsW + i) = *(const float4*)(W + i);
#endif
  __syncthreads();

  int wave = threadIdx.x >> 5, lane = threadIdx.x & 31;
  int m = lane & 15, kh = lane >> 4;
  int tile = blockIdx.x * 8 + wave;
  if (tile > NH/16 - 1) tile = NH/16 - 1;
  int rowbase = tile * 16;

  const float* arow = hx + ((size_t)rowbase + m) * 64;
  __builtin_prefetch(arow + 64*16, 0, 1);
  v2f afrag[16];
  #pragma unroll
  for (int kk = 0; kk < 16; ++kk)
    afrag[kk] = *(const v2f*)(arow + kk*4 + kh*2);

  int ntiles = ldw >> 4;
  for (int t = 0; t < ntiles; ++t) {
    v8f acc = {};
    int col = t*16 + m;
    #pragma unroll
    for (int kk = 0; kk < 16; ++kk) {
      int kc = kk*4 + kh*2;
      v2f b;
      b.x = ldsW[kc*ldw + col];
      b.y = ldsW[(kc+1)*ldw + col];
      acc = __builtin_amdgcn_wmma_f32_16x16x4_f32(false, afrag[kk], false, b,
                                                  (short)0, acc, false, false);
    }
    float bv = bias[col];
    #pragma unroll
    for (int i = 0; i < 8; ++i)
      outp[(size_t)(rowbase + i + kh*8)*ldw + col] = acc[i] + bv;
  }
}

// ============================================================
// Edge message pass: gate = tanh([hd_i|hs_j] . Wa + ba); atomic segment-sum
// One wave32 per edge; 2 channels per lane.
// ============================================================
__global__ void __launch_bounds__(256) k_edge(
    const float* __restrict__ hs, const float* __restrict__ hd,
    const int* __restrict__ ei, const float* __restrict__ Wa,
    const float* __restrict__ ba, float* __restrict__ agg,
    float* __restrict__ cnt)
{
  int wave = threadIdx.x >> 5, lane = threadIdx.x & 31;
  int eg = blockIdx.x * 8 + wave;              // exactly RR*EE waves launched
  int r = eg / EE, e = eg - r * EE;
  int src = ei[((size_t)r*2    )*EE + e];
  int dst = ei[((size_t)r*2 + 1)*EE + e];
  const float* pj = hs + ((size_t)r*NP + src) * 64;
  const float* pi = hd + ((size_t)r*NP + dst) * 64;
  const float* wa = Wa + r * 128;
  float hj0 = pj[lane], hj1 = pj[lane+32];
  float part = pi[lane]*wa[lane] + pi[lane+32]*wa[lane+32]
             + hj0*wa[64+lane]   + hj1*wa[96+lane];
  #pragma unroll
  for (int o = 16; o > 0; o >>= 1) part += __shfl_xor(part, o);
  float gate = tanhf(part + ba[r]);
  float* pa = agg + ((size_t)r*NP + dst) * 64;
  atomicAdd(pa + lane,      gate * hj0);
  atomicAdd(pa + lane + 32, gate * hj1);
  if (lane == 0) atomicAdd(cnt + r*NP + dst, 1.0f);
}

// rel_out = agg / max(cnt,1);  x_next = relu(sum_{r: tdst==t} rel_out / 5)
__global__ void k_combine(const float* __restrict__ agg, const float* __restrict__ cnt,
                          float* __restrict__ xout, int finalLayer)
{
  int idx = blockIdx.x * 256 + threadIdx.x;   // < 3*NP*64
  int c = idx & 63;
  int n = (idx >> 6) % NP;
  int t = idx / (NP * 64);
  float s = 0.f;
  #pragma unroll
  for (int r = 0; r < RR; ++r) {
    if (c_tdst[r] == t) {
      float cc = cnt[r*NP + n];
      float dn = cc > 1.f ? cc : 1.f;
      s += agg[((size_t)r*NP + n)*64 + c] / dn;
    }
  }
  s *= 0.2f;
  s = s > 0.f ? s : 0.f;
  int row = (finalLayer ? c_off[t] : t*NP) + n;   // final layer writes hx layout
  xout[(size_t)row*64 + c] = s;
}

// score[e,h] = q[hdst,h].k[hsrc,h]/8 ; running segment-max via atomicMax on keys
__global__ void __launch_bounds__(256) k_score(
    const float* __restrict__ q, const float* __restrict__ k,
    const int* __restrict__ ei, float* __restrict__ score,
    unsigned* __restrict__ mkey)
{
  int wave = threadIdx.x >> 5, lane = threadIdx.x & 31;
  int eg = blockIdx.x * 8 + wave;
  int r = eg / EE, e = eg - r * EE;
  int src = ei[((size_t)r*2    )*EE + e];
  int dst = ei[((size_t)r*2 + 1)*EE + e];
  int hs_row = src + c_off[c_tsrc[r]];
  int hd_row = dst + c_off[c_tdst[r]];
  const float* qp = q + (size_t)hd_row * 128;
  const float* kp = k + (size_t)hs_row * 128;
  float s0 = qp[lane]    * kp[lane]    + qp[lane+32] * kp[lane+32];
  float s1 = qp[lane+64] * kp[lane+64] + qp[lane+96] * kp[lane+96];
  #pragma unroll
  for (int o = 16; o > 0; o >>= 1) { s0 += __shfl_xor(s0, o); s1 += __shfl_xor(s1, o); }
  if (lane == 0) {
    s0 *= 0.125f; s1 *= 0.125f;
    score[(size_t)eg*2 + 0] = s0;
    score[(size_t)eg*2 + 1] = s1;
    atomicMax(mkey + (size_t)hd_row*2 + 0, fkey(s0));
    atomicMax(mkey + (size_t)hd_row*2 + 1, fkey(s1));
  }
}

// alpha = exp(score - m); den += alpha  (alpha overwrites score buffer)
__global__ void k_alpha(float* __restrict__ score, const unsigned* __restrict__ mkey,
                        const int* __restrict__ ei, float* __restrict__ den)
{
  int idx = blockIdx.x * 256 + threadIdx.x;
  if (idx >= RR*EE*2) return;
  int eg = idx >> 1, h = idx & 1;
  int r = eg / EE, e = eg - r * EE;
  int dst = ei[((size_t)r*2 + 1)*EE + e];
  int hd_row = dst + c_off[c_tdst[r]];
  float mv = funkey(mkey[(size_t)hd_row*2 + h]);
  float a = expf(score[idx] - mv);
  score[idx] = a;
  atomicAdd(den + (size_t)hd_row*2 + h, a);
}

// aggv[hdst] += alpha/(den+eps) * v[hsrc]
__global__ void __launch_bounds__(256) k_aggv(
    const float* __restrict__ score, const float* __restrict__ den,
    const float* __restrict__ v, const int* __restrict__ ei,
    float* __restrict__ aggv)
{
  int wave = threadIdx.x >> 5, lane = threadIdx.x & 31;
  int eg = blockIdx.x * 8 + wave;
  int r = eg / EE, e = eg - r * EE;
  int src = ei[((size_t)r*2    )*EE + e];
  int dst = ei[((size_t)r*2 + 1)*EE + e];
  int hs_row = src + c_off[c_tsrc[r]];
  int hd_row = dst + c_off[c_tdst[r]];
  float a0 = score[(size_t)eg*2 + 0] / (den[(size_t)hd_row*2 + 0] + 1e-16f);
  float a1 = score[(size_t)eg*2 + 1] / (den[(size_t)hd_row*2 + 1] + 1e-16f);
  const float* vp = v + (size_t)hs_row * 128;
  float* op = aggv + (size_t)hd_row * 128;
  atomicAdd(op + lane,      a0 * vp[lane]);
  atomicAdd(op + lane + 32, a0 * vp[lane+32]);
  atomicAdd(op + lane + 64, a1 * vp[lane+64]);
  atomicAdd(op + lane + 96, a1 * vp[lane+96]);
}

// out = mean over heads + skip
__global__ void k_out(const float* __restrict__ aggv, const float* __restrict__ skip,
                      float* __restrict__ outb)
{
  int idx = blockIdx.x * 256 + threadIdx.x;   // < NH*64
  int c = idx & 63, row = idx >> 6;
  outb[idx] = 0.5f * (aggv[(size_t)row*128 + c] + aggv[(size_t)row*128 + 64 + c]) + skip[idx];
}

// per-channel sum / sumsq over 60000 rows (one block per channel)
__global__ void __launch_bounds__(256) k_stats(const float* __restrict__ outb,
                                               float* __restrict__ stats)
{
  __shared__ float ss[256], sq[256];
  int c = blockIdx.x;
  float s = 0.f, q2 = 0.f;
  for (int row = threadIdx.x; row < NH; row += 256) {
    float v = outb[(size_t)row*64 + c];
    s += v; q2 += v * v;
  }
  ss[threadIdx.x] = s; sq[threadIdx.x] = q2;
  __syncthreads();
  for (int o = 128; o > 0; o >>= 1) {
    if (threadIdx.x < o) { ss[threadIdx.x] += ss[threadIdx.x+o]; sq[threadIdx.x] += sq[threadIdx.x+o]; }
    __syncthreads();
  }
  if (threadIdx.x == 0) { stats[c] = ss[0]; stats[64 + c] = sq[0]; }
}

// layernorm(channel-wise over nodes) + leaky relu + output permutation
__global__ void k_norm(const float* __restrict__ outb, const float* __restrict__ stats,
                       const float* __restrict__ gamma, const float* __restrict__ beta,
                       float* __restrict__ dout)
{
  int idx = blockIdx.x * 256 + threadIdx.x;   // < NH*64
  int c = idx & 63, row = idx >> 6;
  const float inv = 1.0f / 60000.0f;
  float mu  = stats[c] * inv;
  float var = stats[64 + c] * inv - mu * mu;
  float y = gamma[c] * (outb[idx] - mu) * rsqrtf(var + 1e-5f) + beta[c];
  y = y >= 0.f ? y : 0.01f * y;
  // hx rows: [audio | visual(type2) | text(type1)] -> out stack [audio, visual(2N..), text(N..)]
  int orow = row < NP ? row : (row < 2*NP ? row + NP : row - NP);
  dout[(size_t)orow*64 + c] = y;
}

__global__ void k_fillf(float* p, long n, float v) {
  long i = (long)blockIdx.x * 256 + threadIdx.x; if (i < n) p[i] = v;
}
__global__ void k_fillu(unsigned* p, long n, unsigned v) {
  long i = (long)blockIdx.x * 256 + threadIdx.x; if (i < n) p[i] = v;
}

extern "C" void kernel_launch(void* const* d_in, const int* in_sizes, int n_in,
                              void* d_out, int out_size, void* d_ws, size_t ws_size,
                              hipStream_t stream)
{
  const float* x     = (const float*)d_in[0];
  const int*   ei    = (const int*)  d_in[1];
  const float* Wlin  = (const float*)d_in[2];
  const float* blin  = (const float*)d_in[3];
  const float* Wa    = (const float*)d_in[4];
  const float* ba    = (const float*)d_in[5];
  const float* Wq    = (const float*)d_in[6];
  const float* bq    = (const float*)d_in[7];
  const float* Wk    = (const float*)d_in[8];
  const float* bk    = (const float*)d_in[9];
  const float* Wv    = (const float*)d_in[10];
  const float* bv    = (const float*)d_in[11];
  const float* Wsk   = (const float*)d_in[12];
  const float* bsk   = (const float*)d_in[13];
  const float* gamma = (const float*)d_in[14];
  const float* beta  = (const float*)d_in[15];

  float* f = (float*)d_ws;
  size_t o = 0;
  float* hsb = f + o; o += (size_t)RR*NP*64;     // 19.2M floats
  float* hdb = f + o; o += (size_t)RR*NP*64;
  float* agg = f + o; o += (size_t)RR*NP*64;
  float* cnt = f + o; o += (size_t)RR*NP;
  float* xd0 = f + o; o += (size_t)3*NP*64;
  float* hx  = f + o; o += (size_t)3*NP*64;
  unsigned* mkey = (unsigned*)(f + o); o += (size_t)NH*2;
  float* den   = f + o; o += (size_t)NH*2;
  float* stats = f + o; o += 128;

  // phase-2 aliases into the (now-free) layer buffers
  float* q     = hsb;
  float* kbuf  = hsb + (size_t)NH*128;
  float* vbuf  = hdb;
  float* skip  = hdb + (size_t)NH*128;
  float* aggv  = agg;
  float* score = agg + (size_t)NH*128;
  float* outb  = score + (size_t)RR*EE*2;

  dim3 blk(256);
  const long aggN = (long)RR*NP*64;

  for (int l = 0; l < 2; ++l) {
    const float* xin = l ? xd0 : x;
    float* xout      = l ? hx  : xd0;
    k_fillf<<<(int)((aggN + 255)/256), blk, 0, stream>>>(agg, aggN, 0.f);
    k_fillf<<<(RR*NP + 255)/256, blk, 0, stream>>>(cnt, (long)RR*NP, 0.f);
    k_lin<<<dim3(157, 30), blk, 0, stream>>>(xin, Wlin + (size_t)l*RR*64*64,
                                             blin + (size_t)l*RR*64, hsb, hdb);
    k_edge<<<RR*EE/8, blk, 0, stream>>>(hsb, hdb, ei, Wa + (size_t)l*RR*128,
                                        ba + (size_t)l*RR, agg, cnt);
    k_combine<<<3*NP*64/256, blk, 0, stream>>>(agg, cnt, xout, l);
  }

  k_qkv<<<dim3(469, 4), blk, 0, stream>>>(hx, Wq, bq, Wk, bk, Wv, bv, Wsk, bsk,
                                          q, kbuf, vbuf, skip);
  k_fillu<<<(NH*2 + 255)/256, blk, 0, stream>>>(mkey, (long)NH*2, 0x00800000u); // key(-FLT_MAX)
  k_fillf<<<(NH*2 + 255)/256, blk, 0, stream>>>(den, (long)NH*2, 0.f);
  k_fillf<<<(int)(((long)NH*128 + 255)/256), blk, 0, stream>>>(aggv, (long)NH*128, 0.f);
  k_score<<<RR*EE/8, blk, 0, stream>>>(q, kbuf, ei, score, mkey);
  k_alpha<<<(RR*EE*2 + 255)/256, blk, 0, stream>>>(score, mkey, ei, den);
  k_aggv<<<RR*EE/8, blk, 0, stream>>>(score, den, vbuf, ei, aggv);
  k_out<<<NH*64/256, blk, 0, stream>>>(aggv, skip, outb);
  k_stats<<<64, blk, 0, stream>>>(outb, stats);
  k_norm<<<NH*64/256, blk, 0, stream>>>(outb, stats, gamma, beta, (float*)d_out);
}